// RepresentationFunc_43164421325168
// MI455X (gfx1250) — compile-verified
//
#include <hip/hip_runtime.h>
#include <hip/hip_bf16.h>

// ---------------------------------------------------------------------------
// 2-layer GCN for MI455X (gfx1250, wave32).
// GEMMs use V_WMMA_F32_16X16X4_F32 (full fp32 precision, 16 K-steps for K=64).
// Edge aggregation uses coalesced gathers + global_atomic_add_f32 (L2-resident:
// whole working set < 192 MB L2).
// ---------------------------------------------------------------------------

typedef __attribute__((ext_vector_type(2))) float v2f;
typedef __attribute__((ext_vector_type(8))) float v8f;

#define NODES_F 64  // feature width (in = hid = out = 64)

// ---------------- degree kernels ----------------
__global__ void k_deg_init(float* __restrict__ deg, int n) {
    int i = blockIdx.x * blockDim.x + threadIdx.x;
    if (i < n) deg[i] = 1.0f;  // self loop contributes 1 to degree
}

__global__ void k_deg_accum(const int* __restrict__ ei, float* __restrict__ deg, int nEdges) {
    int e = blockIdx.x * blockDim.x + threadIdx.x;
    if (e < nEdges) atomicAdd(&deg[ei[e]], 1.0f);  // row = ei[0][e]
}

__global__ void k_deg_rsqrt(float* __restrict__ deg, int n) {
    int i = blockIdx.x * blockDim.x + threadIdx.x;
    if (i < n) deg[i] = rsqrtf(deg[i]);  // dis = deg^-0.5
}

// ---------------- WMMA GEMM: H[n][64] = X[n][64] @ W^T + bias ----------------
// W is row-major [64 out][64 in]  =>  H[m][o] = sum_k X[m][k] * W[o][k] + b[o]
// One wave computes a 16(M) x 64(N) tile with 4 v8f accumulators.
// f32 16x16x4 layouts (ISA 7.12.2):
//   A (16x4): lane<16 -> M=lane, v0=K, v1=K+1 ; lane>=16 -> M=lane-16, v0=K+2, v1=K+3
//   B (4x16): lane<16 -> N=lane, v0=K, v1=K+1 ; lane>=16 -> N=lane-16, v0=K+2, v1=K+3
//   C/D: vgpr j, lane<16 -> (M=j, N=lane) ; lane>=16 -> (M=j+8, N=lane-16)
__global__ void k_gemm64_wmma(const float* __restrict__ X, const float* __restrict__ W,
                              const float* __restrict__ bias, float* __restrict__ H,
                              int nRows) {
    const int lane    = threadIdx.x & 31;
    const int wave    = threadIdx.x >> 5;
    const int tile    = blockIdx.x * (blockDim.x >> 5) + wave;  // 16-row M tile
    const int mBase   = tile * 16;
    if (mBase >= nRows) return;  // wave-uniform guard: EXEC stays all-ones for WMMA

    const int half   = lane >> 4;   // 0: lanes 0-15, 1: lanes 16-31
    const int l15    = lane & 15;
    const int mOff   = 8 * half;

    const float* __restrict__ xrow = X + (size_t)(mBase + l15) * NODES_F;
    const float* __restrict__ w0   = W + (size_t)(l15 +  0) * NODES_F;
    const float* __restrict__ w1   = W + (size_t)(l15 + 16) * NODES_F;
    const float* __restrict__ w2   = W + (size_t)(l15 + 32) * NODES_F;
    const float* __restrict__ w3   = W + (size_t)(l15 + 48) * NODES_F;

    v8f c0 = {}, c1 = {}, c2 = {}, c3 = {};

#pragma unroll
    for (int k = 0; k < NODES_F; k += 4) {
        const int kk = k + 2 * half;
        v2f a, b0, b1, b2, b3;
        a.x  = xrow[kk]; a.y  = xrow[kk + 1];
        b0.x = w0[kk];   b0.y = w0[kk + 1];
        b1.x = w1[kk];   b1.y = w1[kk + 1];
        b2.x = w2[kk];   b2.y = w2[kk + 1];
        b3.x = w3[kk];   b3.y = w3[kk + 1];
        c0 = __builtin_amdgcn_wmma_f32_16x16x4_f32(false, a, false, b0, (short)0, c0, false, false);
        c1 = __builtin_amdgcn_wmma_f32_16x16x4_f32(false, a, false, b1, (short)0, c1, false, false);
        c2 = __builtin_amdgcn_wmma_f32_16x16x4_f32(false, a, false, b2, (short)0, c2, false, false);
        c3 = __builtin_amdgcn_wmma_f32_16x16x4_f32(false, a, false, b3, (short)0, c3, false, false);
    }

    const float bv0 = bias[l15 +  0];
    const float bv1 = bias[l15 + 16];
    const float bv2 = bias[l15 + 32];
    const float bv3 = bias[l15 + 48];

#pragma unroll
    for (int j = 0; j < 8; ++j) {
        const size_t m = (size_t)(mBase + j + mOff);
        float* __restrict__ hrow = H + m * NODES_F;
        hrow[l15 +  0] = c0[j] + bv0;
        hrow[l15 + 16] = c1[j] + bv1;
        hrow[l15 + 32] = c2[j] + bv2;
        hrow[l15 + 48] = c3[j] + bv3;
    }
}

// ---------------- self-loop init: out[i] = dis[i]^2 * h[i] ----------------
__global__ void k_selfloop_init(const float* __restrict__ H, const float* __restrict__ dis,
                                float* __restrict__ out, int n) {
    int t = blockIdx.x * blockDim.x + threadIdx.x;  // n*16 threads, float4 each
    int node = t >> 4;
    if (node >= n) return;
    int f = (t & 15) * 4;
    float d = dis[node];
    float s = d * d;
    const float4 hv = *(const float4*)(H + (size_t)node * NODES_F + f);
    float4 o;
    o.x = s * hv.x; o.y = s * hv.y; o.z = s * hv.z; o.w = s * hv.w;
    *(float4*)(out + (size_t)node * NODES_F + f) = o;
}

// ---------------- edge scatter: out[col] += dis[row]*dis[col]*h[row] ----------------
__global__ void k_scatter(const int* __restrict__ ei, const float* __restrict__ H,
                          const float* __restrict__ dis, float* __restrict__ out,
                          int nEdges) {
    long long t = (long long)blockIdx.x * blockDim.x + threadIdx.x;  // nEdges*16 threads
    int e = (int)(t >> 4);
    if (e >= nEdges) return;
    int f = ((int)t & 15) * 4;
    int r = ei[e];           // source (row)
    int c = ei[nEdges + e];  // target (col)
    float nrm = dis[r] * dis[c];
    const float4 hv = *(const float4*)(H + (size_t)r * NODES_F + f);
    float* __restrict__ o = out + (size_t)c * NODES_F + f;
    atomicAdd(o + 0, nrm * hv.x);
    atomicAdd(o + 1, nrm * hv.y);
    atomicAdd(o + 2, nrm * hv.z);
    atomicAdd(o + 3, nrm * hv.w);
}

// ---------------- elementwise tanh ----------------
__global__ void k_tanh(const float* __restrict__ in, float* __restrict__ out, int n) {
    int i = blockIdx.x * blockDim.x + threadIdx.x;
    if (i < n) out[i] = tanhf(in[i]);
}

// ---------------------------------------------------------------------------
extern "C" void kernel_launch(void* const* d_in, const int* in_sizes, int n_in,
                              void* d_out, int out_size, void* d_ws, size_t ws_size,
                              hipStream_t stream) {
    const float* x  = (const float*)d_in[0];
    const int*   ei = (const int*)d_in[1];
    const float* W1 = (const float*)d_in[2];
    const float* b1 = (const float*)d_in[3];
    const float* W2 = (const float*)d_in[4];
    const float* b2 = (const float*)d_in[5];
    float* out = (float*)d_out;

    const int N = in_sizes[0] / NODES_F;   // 100000 nodes
    const int E = in_sizes[1] / 2;         // 1600000 edges

    // workspace layout
    char* ws = (char*)d_ws;
    float* dis = (float*)ws;                                  // N floats (0.4 MB)
    float* h   = (float*)(ws + 524288);                       // N*64 floats (25.6 MB)
    float* agg = (float*)(ws + 524288 + (size_t)N * NODES_F * 4); // N*64 floats

    const int TB = 256;
    const int nodeBlk  = (N + TB - 1) / TB;
    const int edgeBlk  = (E + TB - 1) / TB;
    const int featTot  = N * NODES_F;
    const int featBlk  = (featTot + TB - 1) / TB;
    const int nodeF4   = (N * 16 + TB - 1) / TB;           // node x float4 threads
    const long long eF4t = (long long)E * 16;
    const int edgeF4   = (int)((eF4t + TB - 1) / TB);      // edge x float4 threads
    const int tiles    = (N + 15) / 16;
    const int gemmBlk  = (tiles + (TB / 32) - 1) / (TB / 32);

    // ---- degrees / normalization ----
    k_deg_init <<<nodeBlk, TB, 0, stream>>>(dis, N);
    k_deg_accum<<<edgeBlk, TB, 0, stream>>>(ei, dis, E);
    k_deg_rsqrt<<<nodeBlk, TB, 0, stream>>>(dis, N);

    // ---- layer 1 ----
    k_gemm64_wmma  <<<gemmBlk, TB, 0, stream>>>(x, W1, b1, h, N);
    k_selfloop_init<<<nodeF4,  TB, 0, stream>>>(h, dis, agg, N);
    k_scatter      <<<edgeF4,  TB, 0, stream>>>(ei, h, dis, agg, E);
    k_tanh         <<<featBlk, TB, 0, stream>>>(agg, h, featTot);   // h = tanh(agg)

    // ---- layer 2 ----
    k_gemm64_wmma  <<<gemmBlk, TB, 0, stream>>>(h, W2, b2, agg, N); // agg = h@W2^T+b2
    k_selfloop_init<<<nodeF4,  TB, 0, stream>>>(agg, dis, out, N);
    k_scatter      <<<edgeF4,  TB, 0, stream>>>(ei, agg, dis, out, E);
    k_tanh         <<<featBlk, TB, 0, stream>>>(out, out, featTot); // in-place tanh
}